// IRMv1Penalty_68444598829186
// MI455X (gfx1250) — compile-verified
//
#include <hip/hip_runtime.h>
#include <hip/hip_bf16.h>

// IRMv1 penalty, MI455X (gfx1250, wave32).
//
// Roofline: 3 read-once streams (f32 logits, i32 targets, i32 env ids) = 192 MB
// -> HBM-bound at 23.3 TB/s (~8.2 us). Stage 1 streams with NT (non-temporal)
// b128 loads (working set == L2 size, read once -> don't pollute L2), computes
// r = (sigmoid(l)-t)*l with v_exp_f32/v_rcp_f32, and bins exactly in fp32 via
// per-wave-private LDS ds_add_f32 atomics (32 sum bins + 32 count bins per
// wave). Stage 2 reduces the 1024x64 per-block partial matrix with
// V_WMMA_F32_16X16X4_F32 (exact fp32 FMA): A = partials in the documented
// 16x4 fp32 A layout, B = ones (layout-invariant), so D rows are column sums.

typedef float f32x4 __attribute__((ext_vector_type(4)));
typedef int   i32x4 __attribute__((ext_vector_type(4)));
typedef float v2f   __attribute__((ext_vector_type(2)));
typedef float v8f   __attribute__((ext_vector_type(8)));

#define NUM_ENVS 32
#define NB 1024            // stage-1 blocks == rows of partial matrix (ws: 256 KB)
#define NT 256             // 8 waves per block (wave32)
#define WAVES (NT / 32)

__device__ __forceinline__ float fast_sigmoid(float x) {
  // sigmoid(x) = 1 / (1 + exp(-x));  exp(-x) = exp2(-x * log2 e)
  float e = __builtin_amdgcn_exp2f(-x * 1.44269504088896340736f); // v_exp_f32
  return __builtin_amdgcn_rcpf(1.0f + e);                         // v_rcp_f32
}

__global__ __launch_bounds__(NT) void irm_stage1(
    const float* __restrict__ logits,
    const int*   __restrict__ targets,
    const int*   __restrict__ envs,
    float* __restrict__ partials,   // [NB][64]: cols 0..31 = sums, 32..63 = counts
    int n)
{
  __shared__ float bins[WAVES * 64];
  for (int i = threadIdx.x; i < WAVES * 64; i += NT) bins[i] = 0.0f;
  __syncthreads();

  const int wave = threadIdx.x >> 5;          // wave32
  float* sumRow = &bins[wave * 64];
  float* cntRow = sumRow + 32;

  const f32x4* L = (const f32x4*)logits;
  const i32x4* T = (const i32x4*)targets;
  const i32x4* E = (const i32x4*)envs;
  const int nthreads = NB * NT;
  const int nchunks  = n >> 2;

  for (int c = blockIdx.x * NT + threadIdx.x; c < nchunks; c += nthreads) {
    f32x4 lv = __builtin_nontemporal_load(&L[c]);   // global_load_b128 th:NT
    i32x4 tv = __builtin_nontemporal_load(&T[c]);
    i32x4 ev = __builtin_nontemporal_load(&E[c]);
#pragma unroll
    for (int j = 0; j < 4; ++j) {
      float x = lv[j];
      float r = (fast_sigmoid(x) - (float)tv[j]) * x;
      int e = ev[j] & (NUM_ENVS - 1);               // ids in [0,32); mask for safety
      atomicAdd(&sumRow[e], r);                     // ds_add_f32 (per-wave private)
      atomicAdd(&cntRow[e], 1.0f);                  // exact: counts <= 2^24 in f32
    }
  }
  // Scalar tail (n % 4), empty for BATCH = 2^24.
  if (blockIdx.x == 0 && threadIdx.x == 0) {
    for (int i = nchunks << 2; i < n; ++i) {
      float x = logits[i];
      float r = (fast_sigmoid(x) - (float)targets[i]) * x;
      int e = envs[i] & (NUM_ENVS - 1);
      atomicAdd(&sumRow[e], r);
      atomicAdd(&cntRow[e], 1.0f);
    }
  }
  __syncthreads();
  if (threadIdx.x < 64) {
    float a = 0.0f;
#pragma unroll
    for (int w = 0; w < WAVES; ++w) a += bins[w * 64 + threadIdx.x];
    partials[blockIdx.x * 64 + threadIdx.x] = a;
  }
}

// One wave. Reduces partials[NB][64] to column sums with WMMA, then the
// per-env epilogue. A-layout (fp32 16x4, ISA 7.12.2): lanes 0-15 hold row
// M=lane with K=0 (VGPR0), K=1 (VGPR1); lanes 16-31 hold K=2,3. B = ones is
// layout-invariant, so D[m][n] = sum_k A[m][k] for every n; we read column
// N=0 via the documented C layout (lane 0 -> M 0-7, lane 16 -> M 8-15).
__global__ __launch_bounds__(32) void irm_stage2(
    const float* __restrict__ partials, float* __restrict__ out, int nrows)
{
  const int l  = threadIdx.x;       // 0..31, EXEC all ones (WMMA requirement)
  const int m  = l & 15;            // A-matrix row owned by this lane
  const int kh = (l >> 4) << 1;     // K base for this lane half: 0 or 2

  v8f c0 = {}, c1 = {}, c2 = {}, c3 = {};
  v2f ones; ones.x = 1.0f; ones.y = 1.0f;

  for (int kb = 0; kb < nrows; kb += 4) {
    const float* r0 = partials + (size_t)(kb + kh) * 64;  // K = kh
    const float* r1 = r0 + 64;                            // K = kh + 1
    v2f a0, a1, a2, a3;
    a0.x = r0[m];        a0.y = r1[m];         // env sums   0..15
    a1.x = r0[16 + m];   a1.y = r1[16 + m];    // env sums  16..31
    a2.x = r0[32 + m];   a2.y = r1[32 + m];    // env counts 0..15
    a3.x = r0[48 + m];   a3.y = r1[48 + m];    // env counts 16..31
    c0 = __builtin_amdgcn_wmma_f32_16x16x4_f32(false, a0, false, ones, (short)0, c0, false, false);
    c1 = __builtin_amdgcn_wmma_f32_16x16x4_f32(false, a1, false, ones, (short)0, c1, false, false);
    c2 = __builtin_amdgcn_wmma_f32_16x16x4_f32(false, a2, false, ones, (short)0, c2, false, false);
    c3 = __builtin_amdgcn_wmma_f32_16x16x4_f32(false, a3, false, ones, (short)0, c3, false, false);
  }

  __shared__ float sums[32], cnts[32];
  if ((l & 15) == 0) {              // lanes 0 and 16 hold column N=0
    int base = (l >> 4) * 8;        // lane 0: M=0..7, lane 16: M=8..15
#pragma unroll
    for (int r = 0; r < 8; ++r) {
      sums[base + r]      = c0[r];
      sums[16 + base + r] = c1[r];
      cnts[base + r]      = c2[r];
      cnts[16 + base + r] = c3[r];
    }
  }
  __syncthreads();

  float cnt = cnts[l];
  float g   = sums[l] / fmaxf(cnt, 1.0f);
  float sq  = (cnt > 0.0f) ? g * g : 0.0f;
  float ne  = (cnt > 0.0f) ? 1.0f  : 0.0f;

  __shared__ float rs[32], rn[32];
  rs[l] = sq; rn[l] = ne;
  __syncthreads();
  if (l == 0) {
    float S = 0.0f, E = 0.0f;
#pragma unroll
    for (int i = 0; i < 32; ++i) { S += rs[i]; E += rn[i]; }
    out[0] = S / fmaxf(E, 1.0f);    // PENALTY_WEIGHT = 1.0
  }
}

extern "C" void kernel_launch(void* const* d_in, const int* in_sizes, int n_in,
                              void* d_out, int out_size, void* d_ws, size_t ws_size,
                              hipStream_t stream) {
  const float* logits  = (const float*)d_in[0];
  const int*   targets = (const int*)d_in[1];
  const int*   envs    = (const int*)d_in[2];
  float* out      = (float*)d_out;
  float* partials = (float*)d_ws;   // needs NB*64*4 = 256 KB of workspace
  const int n = in_sizes[0];

  irm_stage1<<<NB, NT, 0, stream>>>(logits, targets, envs, partials, n);
  irm_stage2<<<1, 32, 0, stream>>>(partials, out, NB);
}